// SpatialTransformer_range_flow_accumulate_flow_85555748536619
// MI455X (gfx1250) — compile-verified
//
#include <hip/hip_runtime.h>
#include <cstdint>

// ---------------------------------------------------------------------------
// src (1,1,96,160,160) f32, flows (3,1,3,96,160,160) f32, range_flow scalar,
// out (1,1,96,160,160) f32.  Memory-bound gather (~170MB @ 23.3TB/s ~ 7us);
// no matrix ops -> no WMMA.  CDNA5 paths: TDM tensor_load_to_lds (TENSORcnt)
// for the flow-tile staging, global_store_async_from_lds_b32 (ASYNCcnt) for
// the transposed output store, ds gathers, split wait counters.
// ---------------------------------------------------------------------------

#define D_ 96
#define H_ 160
#define W_ 160
#define N_ (D_ * H_ * W_)          // 2457600 elements per channel
#define HW_ (H_ * W_)

// Staging tile extents for the flow resample:
//  - W-axis index xi = floor(z_out * 159/96): span over 32 z <= 53  -> 56
//  - H-axis index yi = floor(y  * 159/160): 2 values
//  - D-axis index zi = floor(x_out * 95/160): span over 8 x <= 6    -> 8
#define NXT 56
#define NYT 2
#define NZT 8
#define STAGE_TOT (3 * NZT * NYT * NXT)   // 2688 floats = 10.5 KB
#define SIDX(c, dz, dy, dx) (((((c) * NZT + (dz)) * NYT + (dy)) * NXT) + (dx))

typedef uint32_t u32x4 __attribute__((ext_vector_type(4)));
typedef uint32_t u32x8 __attribute__((ext_vector_type(8)));

// ---- CDNA5 TDM: one DMA op stages the whole 4D tile into LDS --------------
__device__ __forceinline__ void tdm_load(const u32x4& g0, const u32x8& g1,
                                         const u32x4& g2, const u32x4& g3) {
    asm volatile("tensor_load_to_lds %0, %1, %2, %3"
                 :
                 : "s"(g0), "s"(g1), "s"(g2), "s"(g3)
                 : "memory");
}
__device__ __forceinline__ void wait_tensor0() {
#if defined(__has_builtin)
#if __has_builtin(__builtin_amdgcn_s_wait_tensorcnt)
    __builtin_amdgcn_s_wait_tensorcnt(0);
#else
    asm volatile("s_wait_tensorcnt 0x0" ::: "memory");
#endif
#else
    asm volatile("s_wait_tensorcnt 0x0" ::: "memory");
#endif
}

// ---- CDNA5 async store: LDS -> global via the ASYNCcnt engine -------------
__device__ __forceinline__ void async_l2g_b32(float* gptr, const float* lptr) {
    // global_store_async_from_lds_b32 vaddr(64b), vsrc(lds byte addr), off
    asm volatile("global_store_async_from_lds_b32 %0, %1, off"
                 :
                 : "v"((uint64_t)(uintptr_t)gptr),
                   "v"((uint32_t)(uintptr_t)lptr)
                 : "memory");
}

// ---------------------------------------------------------------------------
// Kernel A: fsum[c] = sum over the 3 flow stages.  Pure streaming, float4
// vectorized.  Valid because trilinear sampling is linear and all 3 stages
// are sampled at identical coordinates.
// ---------------------------------------------------------------------------
__global__ __launch_bounds__(256) void flow_sum_kernel(
    const float* __restrict__ flows, float* __restrict__ fsum) {
    const int i = blockIdx.x * 256 + threadIdx.x;       // float4 index
    const float4* f = (const float4*)flows;
    const float4 a = f[i];
    const float4 b = f[i + (3 * N_ / 4)];
    const float4 c = f[i + (6 * N_ / 4)];
    float4 s;
    s.x = a.x + b.x + c.x;
    s.y = a.y + b.y + c.y;
    s.z = a.z + b.z + c.z;
    s.w = a.w + b.w + c.w;
    ((float4*)fsum)[i] = s;
}

// ---------------------------------------------------------------------------
// Kernel B: fused warp + sample.  threadIdx.x = z-in-tile (32), threadIdx.y =
// x-in-tile (8), blockIdx = (z-tile, x-tile, y).  The sampled tensors' fast
// axis (W) tracks output z (the reference feeds channel0/z into the sampler's
// x slot), so this mapping coalesces both gather phases; output stores are
// transposed through LDS and written by the async engine.
// ---------------------------------------------------------------------------
template <bool PRESUMMED>
__global__ __launch_bounds__(256) void warp_sample_kernel(
    const float* __restrict__ src, const float* __restrict__ fbase,
    const float* __restrict__ rfp, float* __restrict__ out) {
    __shared__ float stage[STAGE_TOT];
    __shared__ float otile[32 * 9];     // padded to dodge bank conflicts

    const int tz  = threadIdx.x;                 // 0..31
    const int txo = threadIdx.y;                 // 0..7
    const int tid = tz + 32 * txo;               // 0..255
    const int z0  = blockIdx.x * 32;
    const int x0  = blockIdx.y * 8;
    const int y   = blockIdx.z;

    // Axis scale factors implied by the reference's channel ordering.
    const float SXW = (float)(W_ - 1) / (float)D_;        // z_out -> W axis
    const float SYH = (float)(H_ - 1) / (float)H_;        // y     -> H axis
    const float SZD = (float)(D_ - 1) / (float)W_;        // x_out -> D axis
    const float PXS = (float)(W_ - 1) / (float)(D_ - 1);  // final ch0 -> x px
    const float PZS = (float)(D_ - 1) / (float)(W_ - 1);  // final ch2 -> z px

    // Uniform staging bases (same float expressions as per-lane -> monotone).
    const int xiBase = (int)floorf((float)z0 * SXW);
    const int yiBase = (int)floorf((float)y * SYH);
    const int ziBase = (int)floorf((float)x0 * SZD);

    if (PRESUMMED) {
        // ---- TDM: stage [c:3][zi:8][yi:2][xi:56] tile with one DMA op -----
        if (txo == 0) {   // one wave issues; TENSORcnt is per-wave
            const uint64_t gaddr = (uint64_t)(uintptr_t)
                (fbase + ((size_t)(ziBase * H_ + yiBase) * W_ + xiBase));
            // D# group 0: count=1 | lds_addr | global_addr | type=2
            u32x4 g0 = {1u,
                        (uint32_t)(uintptr_t)&stage[0],
                        (uint32_t)gaddr,
                        (((uint32_t)(gaddr >> 32)) & 0x01FFFFFFu) | (2u << 30)};
            // D# group 1: data_size=4B; remaining extents from tile origin so
            // OOB zero-fill coincides with the real tensor boundary; tile
            // dims 56x2x8; strides: dim0(W)=160, dim1(z-planes)=25600.
            const uint32_t td0 = (uint32_t)(W_ - xiBase);
            const uint32_t td1 = (uint32_t)(H_ - yiBase);
            u32x8 g1 = {2u << 16,
                        (td0 & 0xFFFFu) << 16,
                        (td0 >> 16) | ((td1 & 0xFFFFu) << 16),
                        (td1 >> 16) | ((uint32_t)NXT << 16),
                        (uint32_t)NYT | ((uint32_t)NZT << 16),
                        (uint32_t)W_,
                        ((uint32_t)HW_ & 0xFFFFu) << 16,
                        ((uint32_t)HW_) >> 16};
            // D# group 2: tensor_dim2 = D extent, tensor_dim3 = 3 channels,
            // tensor_dim2_stride = N (channel stride), tile_dim3 = 3.
            u32x4 g2 = {(uint32_t)(D_ - ziBase), 3u, (uint32_t)N_, 3u << 16};
            // D# group 3: dims 4+ unused.
            u32x4 g3 = {0u, 0u, 0u, 0u};
            tdm_load(g0, g1, g2, g3);
            wait_tensor0();
        }
    } else {
        // ---- No workspace: fold the 3-stage sum into plain staging --------
        for (int e = tid; e < STAGE_TOT; e += 256) {
            int dx = e % NXT;
            int r  = e / NXT;
            int dy = r % NYT;
            int r2 = r / NYT;
            int dz = r2 % NZT;
            int c  = r2 / NZT;
            int gx = min(xiBase + dx, W_ - 1);
            int gy = min(yiBase + dy, H_ - 1);
            int gz = min(ziBase + dz, D_ - 1);
            int gi = (gz * H_ + gy) * W_ + gx + c * N_;
            stage[e] = fbase[gi] + fbase[gi + 3 * N_] + fbase[gi + 6 * N_];
        }
    }
    __syncthreads();

    const int z = z0 + tz;
    const int x = x0 + txo;
    const float rf = *rfp;

    // ---- Trilinear resample of summed flows (border padding) --------------
    const float sx = (float)z * SXW;             // in [0, W-1] by construction
    const float sy = (float)y * SYH;
    const float sz = (float)x * SZD;
    int xlo = (int)floorf(sx);  const float fx = sx - (float)xlo;
    int ylo = (int)floorf(sy);  const float fy = sy - (float)ylo;
    int zlo = (int)floorf(sz);  const float fz = sz - (float)zlo;
    const int xhi = min(xlo + 1, W_ - 1);
    const int yhi = min(ylo + 1, H_ - 1);
    const int zhi = min(zlo + 1, D_ - 1);

    const int lx0 = min(max(xlo - xiBase, 0), NXT - 1);
    const int lx1 = min(max(xhi - xiBase, 0), NXT - 1);
    const int ly0 = min(max(ylo - yiBase, 0), NYT - 1);
    const int ly1 = min(max(yhi - yiBase, 0), NYT - 1);
    const int lz0 = min(max(zlo - ziBase, 0), NZT - 1);
    const int lz1 = min(max(zhi - ziBase, 0), NZT - 1);

    const float wx0 = 1.f - fx, wx1 = fx;
    const float wy0 = 1.f - fy, wy1 = fy;
    const float wz0 = 1.f - fz, wz1 = fz;

    float acc[3];
#pragma unroll
    for (int c = 0; c < 3; ++c) {
        float v000 = stage[SIDX(c, lz0, ly0, lx0)];
        float v001 = stage[SIDX(c, lz0, ly0, lx1)];
        float v010 = stage[SIDX(c, lz0, ly1, lx0)];
        float v011 = stage[SIDX(c, lz0, ly1, lx1)];
        float v100 = stage[SIDX(c, lz1, ly0, lx0)];
        float v101 = stage[SIDX(c, lz1, ly0, lx1)];
        float v110 = stage[SIDX(c, lz1, ly1, lx0)];
        float v111 = stage[SIDX(c, lz1, ly1, lx1)];
        acc[c] = wz0 * (wy0 * (wx0 * v000 + wx1 * v001) +
                        wy1 * (wx0 * v010 + wx1 * v011)) +
                 wz1 * (wy0 * (wx0 * v100 + wx1 * v101) +
                        wy1 * (wx0 * v110 + wx1 * v111));
    }

    // final = grid + acc*range_flow; normalize/denormalize collapses to scales
    const float f0 = (float)z + acc[0] * rf;   // channel0 -> sampler x slot
    const float f1 = (float)y + acc[1] * rf;   // channel1 -> sampler y slot
    const float f2 = (float)x + acc[2] * rf;   // channel2 -> sampler z slot
    const float px = f0 * PXS;
    const float py = f1;
    const float pz = f2 * PZS;

    // ---- Trilinear sample of src (zeros padding) --------------------------
    const float pxf = floorf(px), pyf = floorf(py), pzf = floorf(pz);
    const int ix = (int)pxf, iy = (int)pyf, iz = (int)pzf;
    const float gx = px - pxf, gy = py - pyf, gz = pz - pzf;

    float res = 0.f;
#pragma unroll
    for (int dz2 = 0; dz2 < 2; ++dz2) {
#pragma unroll
        for (int dy2 = 0; dy2 < 2; ++dy2) {
#pragma unroll
            for (int dx2 = 0; dx2 < 2; ++dx2) {
                const int cz = iz + dz2, cy = iy + dy2, cx = ix + dx2;
                const bool valid = (cz >= 0) & (cz < D_) & (cy >= 0) &
                                   (cy < H_) & (cx >= 0) & (cx < W_);
                const float w = (dz2 ? gz : 1.f - gz) *
                                (dy2 ? gy : 1.f - gy) *
                                (dx2 ? gx : 1.f - gx);
                const int czc = min(max(cz, 0), D_ - 1);
                const int cyc = min(max(cy, 0), H_ - 1);
                const int cxc = min(max(cx, 0), W_ - 1);
                const float v = src[(czc * H_ + cyc) * W_ + cxc];
                res += valid ? w * v : 0.f;
            }
        }
    }

    // ---- Transpose through LDS; async engine streams LDS -> global --------
    otile[tz * 9 + txo] = res;
    __syncthreads();   // commits ds_store before the async engine reads LDS
    const int zz = tid >> 3;
    const int xx = tid & 7;
    async_l2g_b32(&out[((z0 + zz) * H_ + y) * W_ + x0 + xx],
                  &otile[zz * 9 + xx]);
    // s_endpgm performs an implicit wait-idle, covering ASYNCcnt drain.
}

// ---------------------------------------------------------------------------
extern "C" void kernel_launch(void* const* d_in, const int* in_sizes, int n_in,
                              void* d_out, int out_size, void* d_ws,
                              size_t ws_size, hipStream_t stream) {
    (void)in_sizes; (void)n_in; (void)out_size;
    const float* src   = (const float*)d_in[0];
    const float* flows = (const float*)d_in[1];
    const float* rfp   = (const float*)d_in[2];   // device scalar
    float* out = (float*)d_out;

    const dim3 grid(D_ / 32, W_ / 8, H_);         // (3, 20, 160)
    const dim3 block(32, 8);

    const bool presum = (d_ws != nullptr) &&
                        (ws_size >= (size_t)3 * N_ * sizeof(float));
    if (presum) {
        float* fsum = (float*)d_ws;
        flow_sum_kernel<<<(3 * N_ / 4) / 256, 256, 0, stream>>>(flows, fsum);
        warp_sample_kernel<true><<<grid, block, 0, stream>>>(src, fsum, rfp, out);
    } else {
        warp_sample_kernel<false><<<grid, block, 0, stream>>>(src, flows, rfp, out);
    }
}